// DECModel_68204080660920
// MI455X (gfx1250) — compile-verified
//
#include <hip/hip_runtime.h>

// DEC soft assignment (Student-t, alpha=1) + target distribution.
//   q[i][j] = (1+||z_i-mu_j||^2)^-1, row-normalized
//   p[i][j] = (q^2 / colsum_j), row-normalized
// Memory-bound: ~166 MB total traffic -> ~7us floor at 23.3 TB/s.
// Distance via ||z||^2 - 2 z.mu + ||mu||^2 with the z.mu GEMM done by
// V_WMMA_F32_16X16X4_F32 (full f32 precision, 16x16 tile, K=4 per op).

#define BATCH   262144
#define LDIM    128
#define NCLUST  10
#define KP      16          // centers padded to 16 columns for WMMA
#define STRIDE  132         // LDS row stride in floats: 132 % 64 == 4 -> conflict-free
#define WAVES   4
#define BLOCK   (WAVES * 32)
#define ROWS_PER_BLOCK (WAVES * 16)         // 64
#define NBLK    (BATCH / ROWS_PER_BLOCK)    // 4096

typedef float v2f __attribute__((ext_vector_type(2)));
typedef float v8f __attribute__((ext_vector_type(8)));

// ---------------------------------------------------------------------------
// Kernel 1: per 16-row tile per wave, compute q (row-normalized) and emit
// deterministic per-block column partial sums.
// ---------------------------------------------------------------------------
__global__ __launch_bounds__(BLOCK) void dec_q_kernel(const float* __restrict__ z,
                                                      const float* __restrict__ cen,
                                                      float* __restrict__ qout,
                                                      float* __restrict__ partial) {
    __shared__ float s_c[KP * STRIDE];            //  8448 B, zero-padded centers
    __shared__ float s_z[WAVES][16 * STRIDE];     // 33792 B, z tiles (one per wave)
    __shared__ float s_wsum[WAVES][16];           // per-wave column sums

    const int tid  = threadIdx.x;
    const int wave = tid >> 5;
    const int lane = tid & 31;
    const int m16  = lane & 15;   // row-within-tile (A) / center index N (B)
    const int hi   = lane >> 4;   // K-pair select within WMMA fragment

    // Stage centers, rows 10..15 zero-padded.
    for (int idx = tid; idx < KP * LDIM; idx += BLOCK) {
        const int r = idx >> 7, col = idx & 127;
        s_c[r * STRIDE + col] = (r < NCLUST) ? cen[r * LDIM + col] : 0.0f;
    }
    // Stage this wave's 16 z rows: coalesced 512B global_load_b128 per row.
    const int tile = blockIdx.x * ROWS_PER_BLOCK + wave * 16;
    for (int r = 0; r < 16; ++r) {
        const float4 val = *(const float4*)(z + (size_t)(tile + r) * LDIM + lane * 4);
        *(float4*)(&s_z[wave][r * STRIDE + lane * 4]) = val;
    }
    __syncthreads();

    // Per-lane norms of row m16: cn == ||mu_N||^2 (N = m16), zn == ||z_row||^2.
    float cn = 0.0f, zn = 0.0f;
#pragma unroll
    for (int t = 0; t < LDIM; t += 4) {
        const float4 cv = *(const float4*)(&s_c[m16 * STRIDE + t]);
        const float4 zv = *(const float4*)(&s_z[wave][m16 * STRIDE + t]);
        cn += cv.x * cv.x + cv.y * cv.y + cv.z * cv.z + cv.w * cv.w;
        zn += zv.x * zv.x + zv.y * zv.y + zv.z * zv.z + zv.w * zv.w;
    }

    // D(16x16) = Z_tile(16x128) * C^T(128x16): 32 x V_WMMA_F32_16X16X4_F32.
    // A fragment: lane (m16, hi) holds float2 at (row=m16, col=k0+2*hi).
    // B fragment mirrors A (B[k][n] = centers[n][k] -> same LDS access shape).
    v8f acc = {};
#pragma unroll
    for (int k0 = 0; k0 < LDIM; k0 += 4) {
        const v2f a = *(const v2f*)(&s_z[wave][m16 * STRIDE + k0 + 2 * hi]);
        const v2f b = *(const v2f*)(&s_c[m16 * STRIDE + k0 + 2 * hi]);
        acc = __builtin_amdgcn_wmma_f32_16x16x4_f32(false, a, false, b,
                                                    (short)0, acc, false, false);
    }

    // acc[v]: lane holds D[M][N], M = v + 8*hi, N = m16.
    const int n = m16;
    float colAcc = 0.0f;
#pragma unroll
    for (int v = 0; v < 8; ++v) {
        const float zrow = __shfl(zn, v + 8 * hi, 32);       // ||z_M||^2
        const float sq   = zrow - 2.0f * acc[v] + cn;        // ||z-mu||^2
        const float qun  = (n < NCLUST) ? 1.0f / (1.0f + sq) : 0.0f;
        // Row sum over the 16-lane half (masks 1,2,4,8 stay within the half).
        float s = qun;
        s += __shfl_xor(s, 1, 32);
        s += __shfl_xor(s, 2, 32);
        s += __shfl_xor(s, 4, 32);
        s += __shfl_xor(s, 8, 32);
        const float q = qun / s;
        if (n < NCLUST)
            qout[(size_t)(tile + v + 8 * hi) * NCLUST + n] = q;
        colAcc += q;                                         // q == 0 for n>=10
    }
    // Combine the two half-wave row groups (same N), fixed order -> deterministic.
    colAcc += __shfl_xor(colAcc, 16, 32);
    if (hi == 0) s_wsum[wave][n] = colAcc;
    __syncthreads();
    if (tid < 16) {
        float s = 0.0f;
#pragma unroll
        for (int w = 0; w < WAVES; ++w) s += s_wsum[w][tid];
        partial[blockIdx.x * 16 + tid] = s;                  // deterministic partial
    }
}

// ---------------------------------------------------------------------------
// Kernel 2: deterministic fixed-order reduction of per-block column partials.
// ---------------------------------------------------------------------------
__global__ __launch_bounds__(256) void dec_colsum_reduce(const float* __restrict__ partial,
                                                         float* __restrict__ colsum) {
    __shared__ float s[16][17];
    const int t = threadIdx.x;
    const int j = t & 15, slice = t >> 4;
    const int per = NBLK / 16;                                // 256 blocks/slice
    float acc = 0.0f;
    for (int b = slice * per; b < (slice + 1) * per; ++b)
        acc += partial[b * 16 + j];
    s[slice][j] = acc;
    __syncthreads();
    if (t < 16) {
        float r = 0.0f;
#pragma unroll
        for (int i = 0; i < 16; ++i) r += s[i][t];
        colsum[t] = r;
    }
}

// ---------------------------------------------------------------------------
// Kernel 3: p = rownorm(q^2 / colsum). One thread per row; q is L2-resident.
// ---------------------------------------------------------------------------
__global__ __launch_bounds__(256) void dec_p_kernel(const float* __restrict__ q,
                                                    const float* __restrict__ colsum,
                                                    float* __restrict__ pout) {
    __shared__ float cs[NCLUST];
    if (threadIdx.x < NCLUST) cs[threadIdx.x] = colsum[threadIdx.x];
    __syncthreads();
    const size_t row = (size_t)blockIdx.x * blockDim.x + threadIdx.x;
    const float* qr = q + row * NCLUST;
    float w[NCLUST];
    float s = 0.0f;
#pragma unroll
    for (int j = 0; j < NCLUST; ++j) {
        const float qq = qr[j];
        const float ww = qq * qq / cs[j];
        w[j] = ww;
        s += ww;
    }
    const float inv = 1.0f / s;
    float* pr = pout + row * NCLUST;
#pragma unroll
    for (int j = 0; j < NCLUST; ++j) pr[j] = w[j] * inv;
}

// ---------------------------------------------------------------------------
extern "C" void kernel_launch(void* const* d_in, const int* in_sizes, int n_in,
                              void* d_out, int out_size, void* d_ws, size_t ws_size,
                              hipStream_t stream) {
    const float* z   = (const float*)d_in[0];   // [BATCH, LDIM]
    const float* cen = (const float*)d_in[1];   // [NCLUST, LDIM]
    float* qout = (float*)d_out;                         // [BATCH, NCLUST]
    float* pout = qout + (size_t)BATCH * NCLUST;         // [BATCH, NCLUST]

    float* partial = (float*)d_ws;                       // [NBLK, 16]
    float* colsum  = partial + (size_t)NBLK * 16;        // [16]

    dec_q_kernel<<<NBLK, BLOCK, 0, stream>>>(z, cen, qout, partial);
    dec_colsum_reduce<<<1, 256, 0, stream>>>(partial, colsum);
    dec_p_kernel<<<BATCH / 256, 256, 0, stream>>>(qout, colsum, pout);
}